// Encoder_6_41669772706013
// MI455X (gfx1250) — compile-verified
//
#include <hip/hip_runtime.h>
#include <stdint.h>

#define TT 192
#define CENC 256
#define BATCH 256

typedef __attribute__((ext_vector_type(2))) float v2f;
typedef __attribute__((ext_vector_type(8))) float v8f;

// ---------------------------------------------------------------------------
// Threefry-2x32 (JAX constants/rounds) for exact reproduction of jax.random
// ---------------------------------------------------------------------------
__device__ __forceinline__ uint32_t rotl32(uint32_t x, int r) {
    return (x << r) | (x >> (32 - r));
}

__device__ void threefry2x32(uint32_t k0, uint32_t k1, uint32_t x0, uint32_t x1,
                             uint32_t& o0, uint32_t& o1) {
    uint32_t ks2 = k0 ^ k1 ^ 0x1BD11BDAu;
    x0 += k0; x1 += k1;
    #define RND(r) { x0 += x1; x1 = rotl32(x1, r); x1 ^= x0; }
    RND(13) RND(15) RND(26) RND(6)
    x0 += k1;  x1 += ks2 + 1u;
    RND(17) RND(29) RND(16) RND(24)
    x0 += ks2; x1 += k0 + 2u;
    RND(13) RND(15) RND(26) RND(6)
    x0 += k0;  x1 += k1 + 3u;
    RND(17) RND(29) RND(16) RND(24)
    x0 += k1;  x1 += ks2 + 4u;
    RND(13) RND(15) RND(26) RND(6)
    x0 += ks2; x1 += k0 + 5u;
    #undef RND
    o0 = x0; o1 = x1;
}

// ---------------------------------------------------------------------------
// Generic pad-transpose: in (KK, CI, CO) -> out (KK, CO, P)
//   out[kk][co][p] = (p < CI) ? in[kk][p][co] : 0
// Used for: layer-0 input (B,257,192)->(B,192,260), conv weights
// (5,Cin,256)->(5,256,CinPad), lstm kernels (256,128)->(128,256).
// ---------------------------------------------------------------------------
__global__ void __launch_bounds__(256) transpose_pad(const float* __restrict__ in,
                                                     float* __restrict__ out,
                                                     int CI, int CO, int P) {
    const int row = blockIdx.x;            // kk*CO + co
    const int kk = row / CO, co = row % CO;
    const float* ib = in + (size_t)kk * CI * CO + co;
    float* ob = out + (size_t)row * P;
    for (int p = threadIdx.x; p < P; p += blockDim.x)
        ob[p] = (p < CI) ? ib[(size_t)p * CO] : 0.0f;
}

// ---------------------------------------------------------------------------
// Conv1D (k=5, SAME) as implicit GEMM with V_WMMA_F32_16X16X4_F32.
// X: (B, T, Cs) K-major (zero-padded to Cs), Wt: (5, 256, Cs), Y: (B, T, 256)
// One wave -> 16(t) x 16(cout) tile; each K-chunk: 2x b64 load + 1 WMMA.
// grid = BATCH * (T/16) * 2, 8 waves/block share the A-tile.
// ---------------------------------------------------------------------------
__global__ void __launch_bounds__(256) conv_wmma(const float* __restrict__ X,
                                                 const float* __restrict__ Wt,
                                                 const float* __restrict__ bias,
                                                 float* __restrict__ Y,
                                                 int Cs) {
    const int lane = threadIdx.x & 31;
    const int wave = threadIdx.x >> 5;
    const int btIdx = blockIdx.x >> 1;
    const int half  = blockIdx.x & 1;
    const int b  = btIdx / (TT / 16);
    const int tt = btIdx % (TT / 16);
    const int t0 = tt * 16;
    const int n0 = (half * 8 + wave) * 16;
    const int m    = lane & 15;            // M row (A) / N col (B,D)
    const int koff = (lane >> 4) * 2;      // K sub-offset 0 or 2
    const int n    = n0 + m;

    v8f acc = {};
    const float* Xb = X + (size_t)b * TT * Cs;

    for (int kk = 0; kk < 5; ++kk) {
        const int t = t0 + m + kk - 2;     // SAME padding
        const bool tok = (t >= 0) && (t < TT);
        const int tc = tok ? t : 0;
        const float msk = tok ? 1.0f : 0.0f;
        const float* xr = Xb + (size_t)tc * Cs + koff;
        const float* wr = Wt + ((size_t)(kk * CENC + n)) * Cs + koff;
        #pragma unroll 4
        for (int c0 = 0; c0 < Cs; c0 += 4) {
            v2f a  = *(const v2f*)(xr + c0);
            a.x *= msk; a.y *= msk;
            v2f bf = *(const v2f*)(wr + c0);
            acc = __builtin_amdgcn_wmma_f32_16x16x4_f32(
                false, a, false, bf, (short)0, acc, false, false);
        }
    }

    const float bn = bias[n];
    const int rbase = (lane < 16) ? 0 : 8;
    float* Yb = Y + (size_t)b * TT * CENC;
    #pragma unroll
    for (int r = 0; r < 8; ++r) {
        Yb[(size_t)(t0 + rbase + r) * CENC + n] = acc[r] + bn;
    }
}

// ---------------------------------------------------------------------------
// Per-sample norm over (C,T), affine over t, ReLU. In-place on (B,T,C).
// ---------------------------------------------------------------------------
__global__ void __launch_bounds__(256) groupnorm_relu(float* __restrict__ Y,
                                                      const float* __restrict__ gamma,
                                                      const float* __restrict__ beta) {
    const int b = blockIdx.x, tid = threadIdx.x;
    float* p = Y + (size_t)b * TT * CENC;
    __shared__ float s1[256], s2[256];
    __shared__ float s_mean, s_rstd;
    float a1 = 0.0f, a2 = 0.0f;
    for (int i = tid; i < CENC * TT; i += 256) {
        float v = p[i]; a1 += v; a2 += v * v;
    }
    s1[tid] = a1; s2[tid] = a2;
    __syncthreads();
    for (int s = 128; s > 0; s >>= 1) {
        if (tid < s) { s1[tid] += s1[tid + s]; s2[tid] += s2[tid + s]; }
        __syncthreads();
    }
    if (tid == 0) {
        const float inv_n = 1.0f / (float)(CENC * TT);
        float mean = s1[0] * inv_n;
        float var  = s2[0] * inv_n - mean * mean;
        s_mean = mean;
        s_rstd = rsqrtf(var + 1e-5f);
    }
    __syncthreads();
    const float mean = s_mean, rstd = s_rstd;
    for (int i = tid; i < CENC * TT; i += 256) {
        int t = i >> 8;                     // (B,T,C): row index = t
        float v = (p[i] - mean) * rstd * gamma[t] + beta[t];
        p[i] = v > 0.0f ? v : 0.0f;
    }
}

// ---------------------------------------------------------------------------
// interp_lnr: random segment time-warp + ragged repack. 1 block / sample.
// in,out: (B, T, C) -- time rows contiguous, fully coalesced gather/scatter.
// ---------------------------------------------------------------------------
__global__ void __launch_bounds__(256) interp_kernel(const float* __restrict__ in,
                                                     float* __restrict__ out,
                                                     int layer) {
    const int b = blockIdx.x, tid = threadIdx.x;
    __shared__ float s_scale[7];
    __shared__ int   s_len[7];
    __shared__ int   s_off[7];
    __shared__ float s_lam[448];
    __shared__ short s_ifl[448];
    __shared__ short s_icl[448];
    __shared__ short s_pos[448];
    __shared__ unsigned char s_mask[448];
    __shared__ int s_cnt;

    if (tid < 7) {
        // key chain: key(42) -> fold_in(layer) -> split into k1 (scales), k2 (lens)
        uint32_t f0, f1;
        threefry2x32(0u, 42u, 0u, (uint32_t)layer, f0, f1);
        uint32_t a0, a1, c0, c1;
        threefry2x32(f0, f1, 0u, 2u, a0, a1);   // split lane 0
        threefry2x32(f0, f1, 1u, 3u, c0, c1);   // split lane 1
        // k1 = (a0, c0), k2 = (a1, c1)
        const int n = b * 7 + tid;              // index into (1792,) random arrays
        const uint32_t ln = (n < 896) ? (uint32_t)n : (uint32_t)(n - 896);
        uint32_t y0, y1;
        // scales = uniform(k1, 0.5..1.5)
        threefry2x32(a0, c0, ln, 896u + ln, y0, y1);
        uint32_t bits = (n < 896) ? y0 : y1;
        float u = __uint_as_float((bits >> 9) | 0x3F800000u) - 1.0f;
        s_scale[tid] = 0.5f + u;
        // len_seg = randint(k2, 19, 32): span 13, multiplier (2^16%13)^2%13 = 9
        threefry2x32(a1, c1, ln, 896u + ln, y0, y1);
        bits = (n < 896) ? y0 : y1;
        uint32_t off = (((bits >> 16) % 13u) * 9u + (bits & 0xFFFFu) % 13u) % 13u;
        s_len[tid] = 19 + (int)off;
    }
    __syncthreads();
    if (tid == 0) {
        int acc = 0;
        for (int s = 0; s < 7; ++s) { s_off[s] = acc; acc += s_len[s]; }
    }
    __syncthreads();
    for (int j = tid; j < 448; j += 256) {
        const int s = j >> 6, i64 = j & 63;
        float isc = (float)i64 / s_scale[s];
        float fl  = floorf(isc);
        float lam = isc - fl;
        bool mseg = fl < (float)(s_len[s] - 1);
        float idx_org = fl + (float)s_off[s];
        bool mlen = idx_org < (float)(TT - 1);
        int ii  = (int)idx_org;
        int ifl = min(max(ii, 0), TT - 1);
        int icl = min(max(ii + 1, 0), TT - 1);
        s_lam[j] = lam;
        s_ifl[j] = (short)ifl;
        s_icl[j] = (short)icl;
        s_mask[j] = (mseg && mlen) ? 1 : 0;
    }
    __syncthreads();
    if (tid == 0) {
        int p = 0;
        for (int j = 0; j < 448; ++j) {
            s_pos[j] = s_mask[j] ? (short)p : (short)-1;
            if (s_mask[j]) ++p;
        }
        s_cnt = p > TT ? TT : p;
    }
    __syncthreads();

    const float* ib = in  + (size_t)b * TT * CENC;
    float*       ob = out + (size_t)b * TT * CENC;
    const int c = tid;                          // 256 threads == 256 channels
    for (int j = 0; j < 448; ++j) {
        const int p = (int)s_pos[j];
        if (p >= 0 && p < TT) {
            const float lam = s_lam[j];
            const float v = (1.0f - lam) * ib[(size_t)s_ifl[j] * CENC + c]
                          +         lam  * ib[(size_t)s_icl[j] * CENC + c];
            ob[(size_t)p * CENC + c] = v;
        }
    }
    for (int t = s_cnt; t < TT; ++t) ob[(size_t)t * CENC + c] = 0.0f;
}

// ---------------------------------------------------------------------------
// LSTM input GEMM with WMMA: XK[b,t,0:128]=x@Kf+bf, XK[b,t,128:256]=x@Kb+bb
// act: (B,T,256), KT: (256,256) fused transposed [Kf^T ; Kb^T], xk: (B,T,256)
// grid = (B*T/16) * 2
// ---------------------------------------------------------------------------
__global__ void __launch_bounds__(256) lstm_in_gemm(const float* __restrict__ act,
                                                    const float* __restrict__ KT,
                                                    const float* __restrict__ bf,
                                                    const float* __restrict__ bb,
                                                    float* __restrict__ xk) {
    const int lane = threadIdx.x & 31;
    const int wave = threadIdx.x >> 5;
    const int mt   = blockIdx.x >> 1;
    const int half = blockIdx.x & 1;
    const int m0 = mt * 16;
    const int b  = m0 / TT;
    const int t0 = m0 % TT;
    const int n0 = (half * 8 + wave) * 16;
    const int m    = lane & 15;
    const int koff = (lane >> 4) * 2;
    const int n    = n0 + m;

    v8f acc = {};
    const float* ar = act + ((size_t)(b * TT) + t0 + m) * CENC + koff;
    const float* wr = KT + (size_t)n * CENC + koff;
    #pragma unroll 4
    for (int k0 = 0; k0 < CENC; k0 += 4) {
        v2f a   = *(const v2f*)(ar + k0);
        v2f bfr = *(const v2f*)(wr + k0);
        acc = __builtin_amdgcn_wmma_f32_16x16x4_f32(
            false, a, false, bfr, (short)0, acc, false, false);
    }
    const float bn = (n < 128) ? bf[n] : bb[n - 128];
    const int rbase = (lane < 16) ? 0 : 8;
    #pragma unroll
    for (int r = 0; r < 8; ++r) {
        const int t = t0 + rbase + r;
        xk[((size_t)(b * TT + t)) * CENC + n] = acc[r] + bn;
    }
}

// ---------------------------------------------------------------------------
// Bi-LSTM recurrence: one wave32 per (sample, direction); lane = hidden unit.
// xk: (B,T,256)  rf,rb: (32,128)  out: (B,T,64)
// ---------------------------------------------------------------------------
__global__ void __launch_bounds__(256) lstm_rec(const float* __restrict__ xk,
                                                const float* __restrict__ rf,
                                                const float* __restrict__ rb,
                                                float* __restrict__ out) {
    __shared__ float s_r[2][32 * 128];
    const int tid = threadIdx.x;
    for (int i = tid; i < 32 * 128; i += 256) {
        s_r[0][i] = rf[i];
        s_r[1][i] = rb[i];
    }
    __syncthreads();
    const int wave = tid >> 5, lane = tid & 31;
    const int task = blockIdx.x * 8 + wave;   // 0..511
    const int dir = task & 1;
    const int b   = task >> 1;
    const float* rr = s_r[dir];
    float h = 0.0f, c = 0.0f;
    for (int step = 0; step < TT; ++step) {
        const int t = dir ? (TT - 1 - step) : step;
        const float* zb = xk + ((size_t)(b * TT + t)) * 256 + dir * 128;
        float zi = zb[lane], zf = zb[32 + lane], zg = zb[64 + lane], zo = zb[96 + lane];
        #pragma unroll 4
        for (int k = 0; k < 32; ++k) {
            const float hk = __shfl(h, k, 32);
            const float* rk = rr + k * 128;
            zi += hk * rk[lane];
            zf += hk * rk[32 + lane];
            zg += hk * rk[64 + lane];
            zo += hk * rk[96 + lane];
        }
        const float ig = 1.0f / (1.0f + expf(-zi));
        const float fg = 1.0f / (1.0f + expf(-zf));
        const float gg = tanhf(zg);
        const float og = 1.0f / (1.0f + expf(-zo));
        c = fg * c + ig * gg;
        h = og * tanhf(c);
        out[((size_t)(b * TT + t)) * 64 + dir * 32 + lane] = h;
    }
}

// ---------------------------------------------------------------------------
extern "C" void kernel_launch(void* const* d_in, const int* in_sizes, int n_in,
                              void* d_out, int out_size, void* d_ws, size_t ws_size,
                              hipStream_t stream) {
    const float* x   = (const float*)d_in[0];
    const float* w0  = (const float*)d_in[1];
    const float* b0  = (const float*)d_in[2];
    const float* w1  = (const float*)d_in[3];
    const float* b1  = (const float*)d_in[4];
    const float* w2  = (const float*)d_in[5];
    const float* b2  = (const float*)d_in[6];
    const float* g0  = (const float*)d_in[7];
    const float* be0 = (const float*)d_in[8];
    const float* g1  = (const float*)d_in[9];
    const float* be1 = (const float*)d_in[10];
    const float* g2  = (const float*)d_in[11];
    const float* be2 = (const float*)d_in[12];
    const float* kf  = (const float*)d_in[13];
    const float* rf  = (const float*)d_in[14];
    const float* bf  = (const float*)d_in[15];
    const float* kb  = (const float*)d_in[16];
    const float* rb  = (const float*)d_in[17];
    const float* bb  = (const float*)d_in[18];
    float* out = (float*)d_out;

    const int C0P = 260;                              // 257 padded to /4, keeps b64 align
    float* wsXT = (float*)d_ws;                       // (B,192,260) = 12,779,520 f
    float* wsY  = wsXT + (size_t)BATCH * TT * C0P;    // (B,192,256) = 12,582,912 f
    float* wsA  = wsY  + (size_t)BATCH * TT * CENC;   // (B,192,256)
    float* wsWT = wsA  + (size_t)BATCH * TT * CENC;   // (5,256,260) = 332,800 f
    float* wsKT = wsWT + (size_t)5 * CENC * C0P;      // (256,256)   = 65,536 f
    float* wsXK = wsXT;                               // reuse after conv0

    const int convGrid = BATCH * (TT / 16) * 2;       // 6144 blocks, 8 waves each

    // layer 0: transpose input (B,257,192)->(B,192,260), weights (5,257,256)->(5,256,260)
    transpose_pad<<<BATCH * TT, 256, 0, stream>>>(x, wsXT, 257, TT, C0P);
    transpose_pad<<<5 * CENC, 256, 0, stream>>>(w0, wsWT, 257, CENC, C0P);
    conv_wmma<<<convGrid, 256, 0, stream>>>(wsXT, wsWT, b0, wsY, C0P);
    groupnorm_relu<<<BATCH, 256, 0, stream>>>(wsY, g0, be0);
    interp_kernel<<<BATCH, 256, 0, stream>>>(wsY, wsA, 0);
    // layer 1
    transpose_pad<<<5 * CENC, 256, 0, stream>>>(w1, wsWT, CENC, CENC, CENC);
    conv_wmma<<<convGrid, 256, 0, stream>>>(wsA, wsWT, b1, wsY, CENC);
    groupnorm_relu<<<BATCH, 256, 0, stream>>>(wsY, g1, be1);
    interp_kernel<<<BATCH, 256, 0, stream>>>(wsY, wsA, 1);
    // layer 2
    transpose_pad<<<5 * CENC, 256, 0, stream>>>(w2, wsWT, CENC, CENC, CENC);
    conv_wmma<<<convGrid, 256, 0, stream>>>(wsA, wsWT, b2, wsY, CENC);
    groupnorm_relu<<<BATCH, 256, 0, stream>>>(wsY, g2, be2);
    interp_kernel<<<BATCH, 256, 0, stream>>>(wsY, wsA, 2);
    // LSTM: fuse Kf^T (rows 0..127) and Kb^T (rows 128..255) into one K-major table
    transpose_pad<<<128, 256, 0, stream>>>(kf, wsKT, CENC, 128, CENC);
    transpose_pad<<<128, 256, 0, stream>>>(kb, wsKT + (size_t)128 * CENC, CENC, 128, CENC);
    lstm_in_gemm<<<convGrid, 256, 0, stream>>>(wsA, wsKT, bf, bb, wsXK);
    lstm_rec<<<(BATCH * 2) / 8, 256, 0, stream>>>(wsXK, rf, rb, out);
}